// CTRule_7421703488266
// MI455X (gfx1250) — compile-verified
//
#include <hip/hip_runtime.h>

typedef __attribute__((ext_vector_type(2))) float v2f;
typedef __attribute__((ext_vector_type(8))) float v8f;

#define RANKC   128
#define HRANK   64
#define NENT    40000
#define BATCH   1024
#define CYCLEC  120
#define COLGRPS 625   // 40000 / 64
#define ROWTILES 64   // 1024 / 16

// ---------------------------------------------------------------------------
// Kernel 1: per-(row, j) elementwise prologue -> V[b][k] = a+b  (1024 x 128)
// ---------------------------------------------------------------------------
__global__ __launch_bounds__(64) void prep_kernel(
    const int*  __restrict__ x,
    const float* __restrict__ E0, const float* __restrict__ E1,
    const float* __restrict__ E2, const float* __restrict__ E3,
    const float* __restrict__ E4, const float* __restrict__ E5,
    const float* __restrict__ E6, const float* __restrict__ rule_C,
    const float* __restrict__ rule_S, const unsigned char* __restrict__ has_rules,
    float* __restrict__ V)
{
  const int b = blockIdx.x;
  const int j = threadIdx.x;           // 0..63

  const int h  = x[b * 4 + 0];
  const int r  = x[b * 4 + 1];
  const int t  = x[b * 4 + 3];
  const int tb = t / CYCLEC;

  const float lhs_a = E0[(size_t)h * RANKC + j];
  const float lhs_b = E0[(size_t)h * RANKC + HRANK + j];
  const float rel_a = E1[(size_t)r * RANKC + j];
  const float rel_b = E1[(size_t)r * RANKC + HRANK + j];
  const float ct_a  = E4[(size_t)t * RANKC + j];
  const float ct_b  = E4[(size_t)t * RANKC + HRANK + j];
  const float tm_a  = E2[(size_t)t * RANKC + j]        + E5[(size_t)tb * RANKC + j];
  const float tm_b  = E2[(size_t)t * RANKC + HRANK + j] + E5[(size_t)tb * RANKC + HRANK + j];
  const float te_a  = E3[(size_t)t * RANKC + j]        + E6[(size_t)tb * RANKC + j];
  const float te_b  = E3[(size_t)t * RANKC + HRANK + j] + E6[(size_t)tb * RANKC + HRANK + j];
  const float rc_a  = rule_C[(size_t)r * RANKC + j];
  const float rc_b  = rule_C[(size_t)r * RANKC + HRANK + j];
  const float s     = rule_S[r];
  const bool  hr    = has_rules[r] != 0;

  // rule_branch = cmul(comp_time, rule_C) - s*rel
  const float rule_a = ct_a * rc_a - ct_b * rc_b - s * rel_a;
  const float rule_b = ct_a * rc_b + ct_b * rc_a - s * rel_b;
  // norule_branch = lhs + cmul(rel, lhs)
  const float noru_a = lhs_a + (rel_a * lhs_a - rel_b * lhs_b);
  const float noru_b = lhs_b + (rel_a * lhs_b + rel_b * lhs_a);
  const float rs_a = hr ? rule_a : noru_a;
  const float rs_b = hr ? rule_b : noru_b;

  // rel_ = rel + complex_mul(rel, comp_time) + complex_mul(rel, rule_score)
  //   complex_mul(e1,e2): (a*c + b*d, a*d - b*c)
  const float rr_a = rel_a + (rel_a * ct_a + rel_b * ct_b) + (rel_a * rs_a + rel_b * rs_b);
  const float rr_b = rel_b + (rel_a * ct_b - rel_b * ct_a) + (rel_a * rs_b - rel_b * rs_a);

  // mul4(e1=[lhs_a,lhs_b,te_a,te_b], e2=[rr_a,rr_b,tm_a,tm_b])
  const float A = lhs_a, Bq = lhs_b, C = te_a, D = te_b;
  const float E = rr_a,  F  = rr_b,  G = tm_a, H = tm_b;
  const float o0 = A * E - Bq * F - C * G - D * H;
  const float o1 = Bq * E + A * F + C * H - D * G;
  const float o2 = C * E + A * G + D * F - Bq * H;
  const float o3 = D * E + A * H + Bq * G - C * F;

  // scores row vector contribution: a_half + b_half
  V[(size_t)b * RANKC + j]          = o0 + o2;
  V[(size_t)b * RANKC + HRANK + j]  = o1 + o3;
}

// ---------------------------------------------------------------------------
// Kernel 2: scores[1024 x 40000] = V[1024 x 128] @ E0^T via v_wmma_f32_16x16x4_f32
// One wave computes a 16x64 output tile (4 accumulators, A reused 4x).
// ---------------------------------------------------------------------------
__global__ __launch_bounds__(256) void gemm_wmma_kernel(
    const float* __restrict__ V,
    const float* __restrict__ E0,
    float* __restrict__ out)
{
  const int lane = threadIdx.x & 31;
  const int wib  = threadIdx.x >> 5;                 // wave in block (0..7)
  const int gwid = blockIdx.x * 8 + wib;             // 0..39999
  const int colg = gwid % COLGRPS;                   // 64-wide column group
  const int rowt = gwid / COLGRPS;                   // 16-row tile

  const int row_base = rowt * 16;
  const int col_base = colg * 64;

  const int half = lane >> 4;                        // K sub-pair selector
  const int m    = lane & 15;                        // M (for A) / N (for B)

  // A fragment source: V[row_base+m][kk + 2*half + {0,1}]
  const float* ap  = V  + (size_t)(row_base + m) * RANKC + 2 * half;
  // B fragment source: B[k][n] = E0[n][k]
  const float* bp0 = E0 + (size_t)(col_base +  0 + m) * RANKC + 2 * half;
  const float* bp1 = E0 + (size_t)(col_base + 16 + m) * RANKC + 2 * half;
  const float* bp2 = E0 + (size_t)(col_base + 32 + m) * RANKC + 2 * half;
  const float* bp3 = E0 + (size_t)(col_base + 48 + m) * RANKC + 2 * half;

  v8f acc0 = {}, acc1 = {}, acc2 = {}, acc3 = {};

#pragma unroll 4
  for (int kk = 0; kk < RANKC; kk += 4) {
    const v2f a  = *(const v2f*)(ap  + kk);
    const v2f b0 = *(const v2f*)(bp0 + kk);
    const v2f b1 = *(const v2f*)(bp1 + kk);
    const v2f b2 = *(const v2f*)(bp2 + kk);
    const v2f b3 = *(const v2f*)(bp3 + kk);
    acc0 = __builtin_amdgcn_wmma_f32_16x16x4_f32(false, a, false, b0, (short)0, acc0, false, false);
    acc1 = __builtin_amdgcn_wmma_f32_16x16x4_f32(false, a, false, b1, (short)0, acc1, false, false);
    acc2 = __builtin_amdgcn_wmma_f32_16x16x4_f32(false, a, false, b2, (short)0, acc2, false, false);
    acc3 = __builtin_amdgcn_wmma_f32_16x16x4_f32(false, a, false, b3, (short)0, acc3, false, false);
  }

  // C/D layout: VGPR g -> row (g + 8*half), col = col_base + tile*16 + m
  float* o = out + (size_t)(row_base) * NENT + col_base + m;
#pragma unroll
  for (int g = 0; g < 8; ++g) {
    const size_t off = (size_t)(g + 8 * half) * NENT;
    __builtin_nontemporal_store(acc0[g], o + off +  0);
    __builtin_nontemporal_store(acc1[g], o + off + 16);
    __builtin_nontemporal_store(acc2[g], o + off + 32);
    __builtin_nontemporal_store(acc3[g], o + off + 48);
  }
}

// ---------------------------------------------------------------------------
extern "C" void kernel_launch(void* const* d_in, const int* in_sizes, int n_in,
                              void* d_out, int out_size, void* d_ws, size_t ws_size,
                              hipStream_t stream) {
  const int*   x        = (const int*)d_in[0];
  const float* E0       = (const float*)d_in[1];
  const float* E1       = (const float*)d_in[2];
  const float* E2       = (const float*)d_in[3];
  const float* E3       = (const float*)d_in[4];
  const float* E4       = (const float*)d_in[5];
  const float* E5       = (const float*)d_in[6];
  const float* E6       = (const float*)d_in[7];
  const float* rule_C   = (const float*)d_in[8];
  const float* rule_S   = (const float*)d_in[9];
  const unsigned char* has_rules = (const unsigned char*)d_in[10];

  float* V   = (float*)d_ws;        // 1024*128 floats = 512 KB
  float* out = (float*)d_out;       // 1024*40000 floats

  prep_kernel<<<BATCH, 64, 0, stream>>>(x, E0, E1, E2, E3, E4, E5, E6,
                                        rule_C, rule_S, has_rules, V);

  // 40000 waves total = 5000 blocks x 8 waves
  gemm_wmma_kernel<<<(COLGRPS * ROWTILES) / 8, 256, 0, stream>>>(V, E0, out);
}